// DualGATConv_75445395522170
// MI455X (gfx1250) — compile-verified
//
#include <hip/hip_runtime.h>
#include <hip/hip_bf16.h>
#include <stdint.h>

// ---------------------------------------------------------------------------
// Dual GATConv for MI455X (gfx1250, wave32).
//   h = x @ W            -> V_WMMA_F32_16X16X4_F32 (fp32-exact matrix path)
//   attention softmax    -> float atomic max/add over destination segments
//   aggregation          -> wave-per-edge float4 + global_atomic_add_f32
// ---------------------------------------------------------------------------

#define IN_F      256
#define HC        128
#define NHEAD     4
#define CPH       32
#define NEG_SLOPE 0.2f

typedef float v2f __attribute__((ext_vector_type(2)));
typedef float v8f __attribute__((ext_vector_type(8)));

__device__ __forceinline__ float lrelu(float x) {
  return x > 0.f ? x : NEG_SLOPE * x;
}

// Mixed-sign float atomic max: int-max for >=0, uint-min for <0. Correct for
// any interleaving because positive payloads always win the int compare and
// negative payloads only shrink already-negative bit patterns.
__device__ __forceinline__ void atomic_max_f32(float* addr, float val) {
  if (val >= 0.f) atomicMax((int*)addr, __float_as_int(val));
  else            atomicMin((unsigned int*)addr, __float_as_uint(val));
}

// ------------------------------------------------------------------ GEMM ----
// Hout[N,128] = X[N,256] * W[256,128], fp32 WMMA (16x16x4), K-loop of 64.
// Block = 256 thr = 8 waves; block computes 16 rows x 128 cols.
#define LDS_STRIDE 260   // 256 + 4 pad -> bank = (4*m + k) % 64, conflict-free

__global__ __launch_bounds__(256) void gemm_xw_wmma(
    const float* __restrict__ X, const float* __restrict__ W,
    float* __restrict__ Hout, int nrows) {
  __shared__ float As[16 * LDS_STRIDE];
  const int tid   = threadIdx.x;
  const int mbase = blockIdx.x * 16;

  // Stage the 16x256 X tile into padded LDS (16 threads per row, float4).
  {
    const int r = tid >> 4;
    const int c = tid & 15;
    int row = mbase + r; if (row >= nrows) row = nrows - 1;
    const float4* Xrow = (const float4*)(X + (size_t)row * IN_F);
    float4* Arow = (float4*)(As + r * LDS_STRIDE);   // 1040B rows stay 16B-aligned
#pragma unroll
    for (int i = 0; i < 4; ++i) Arow[c + 16 * i] = Xrow[c + 16 * i];
  }
  __syncthreads();

  const int wave = tid >> 5;                // 8 waves -> 8 column slabs of 16
  const int lane = tid & 31;
  const int m    = lane & 15;               // A: lanes 0-15 K{0,1}, 16-31 K{2,3}
  const int kb   = (lane >> 4) << 1;
  const int ncol = wave * 16 + (lane & 15); // B/C: N striped across lanes

  v8f acc = {};
  const float* Arow = As + m * LDS_STRIDE;
#pragma unroll 8
  for (int k = 0; k < IN_F; k += 4) {
    v2f a = *(const v2f*)(Arow + k + kb);   // ds_load_b64, conflict-free
    v2f b;
    b.x = W[(size_t)(k + kb)     * HC + ncol];
    b.y = W[(size_t)(k + kb + 1) * HC + ncol];
    acc = __builtin_amdgcn_wmma_f32_16x16x4_f32(false, a, false, b,
                                                (short)0, acc, false, false);
  }

  // C layout: VGPR r -> M = r (lanes 0-15) / 8+r (lanes 16-31), N = lane%16.
  const int rbase = mbase + ((lane >> 4) << 3);
#pragma unroll
  for (int r = 0; r < 8; ++r) {
    const int row = rbase + r;
    if (row < nrows) Hout[(size_t)row * HC + ncol] = acc[r];
  }
}

// ------------------------------------------------------- node-level pass ----
// a_src/a_dst per (node, head); self-loop alpha seeds the segment max.
__global__ void node_att(const float* __restrict__ h,
                         const float* __restrict__ att_src,
                         const float* __restrict__ att_dst,
                         float* __restrict__ asrc, float* __restrict__ adst,
                         float* __restrict__ amax, int N) {
  int t = blockIdx.x * blockDim.x + threadIdx.x;
  if (t >= N * NHEAD) return;
  const int n = t >> 2, hd = t & 3;
  const float* hp = h + (size_t)n * HC + hd * CPH;
  const float* sa = att_src + hd * CPH;
  const float* da = att_dst + hd * CPH;
  float s1 = 0.f, s2 = 0.f;
#pragma unroll
  for (int c = 0; c < CPH; ++c) { float v = hp[c]; s1 += v * sa[c]; s2 += v * da[c]; }
  asrc[t] = s1;
  adst[t] = s2;
  amax[t] = lrelu(s1 + s2);   // self loop always present -> valid max seed
}

// den[n,h] starts from the self-loop term (runs after edge max pass).
__global__ void node_den(const float* __restrict__ asrc,
                         const float* __restrict__ adst,
                         const float* __restrict__ amax,
                         float* __restrict__ den, int N) {
  int t = blockIdx.x * blockDim.x + threadIdx.x;
  if (t >= N * NHEAD) return;
  den[t] = __expf(lrelu(asrc[t] + adst[t]) - amax[t]);
}

// ------------------------------------------------------- edge-level pass ----
__global__ void edge_amax(const long long* __restrict__ ei,
                          const float* __restrict__ asrc,
                          const float* __restrict__ adst,
                          float* __restrict__ amax, int E) {
  int t = blockIdx.x * blockDim.x + threadIdx.x;
  if (t >= E * NHEAD) return;
  const int e = t >> 2, hd = t & 3;
  const int s = (int)ei[e], d = (int)ei[E + e];
  const float al = lrelu(asrc[s * NHEAD + hd] + adst[d * NHEAD + hd]);
  atomic_max_f32(&amax[d * NHEAD + hd], al);
}

__global__ void edge_den(const long long* __restrict__ ei,
                         const float* __restrict__ asrc,
                         const float* __restrict__ adst,
                         const float* __restrict__ amax,
                         float* __restrict__ den, int E) {
  int t = blockIdx.x * blockDim.x + threadIdx.x;
  if (t >= E * NHEAD) return;
  const int e = t >> 2, hd = t & 3;
  const int s = (int)ei[e], d = (int)ei[E + e];
  const float al = lrelu(asrc[s * NHEAD + hd] + adst[d * NHEAD + hd]);
  atomicAdd(&den[d * NHEAD + hd], __expf(al - amax[d * NHEAD + hd]));
}

// Wave-per-edge aggregation: lane l owns features [4l,4l+4) (head = l>>3).
__global__ __launch_bounds__(256) void edge_aggregate(
    const long long* __restrict__ ei, const float* __restrict__ h,
    const float* __restrict__ asrc, const float* __restrict__ adst,
    const float* __restrict__ amax, const float* __restrict__ den,
    float* __restrict__ acc, int E) {
  const int e = blockIdx.x * (blockDim.x >> 5) + (threadIdx.x >> 5);
  if (e >= E) return;
  const int lane = threadIdx.x & 31;
  const int s = (int)ei[e], d = (int)ei[E + e];
  const int hd = lane >> 3;
  const float al = lrelu(asrc[s * NHEAD + hd] + adst[d * NHEAD + hd]);
  const float w  = __expf(al - amax[d * NHEAD + hd]) / den[d * NHEAD + hd];
  const float4 hv = ((const float4*)(h + (size_t)s * HC))[lane];
  float* dst = acc + (size_t)d * HC + lane * 4;
  atomicAdd(dst + 0, hv.x * w);
  atomicAdd(dst + 1, hv.y * w);
  atomicAdd(dst + 2, hv.z * w);
  atomicAdd(dst + 3, hv.w * w);
}

// --------------------------------------------------------------- finalize ---
__global__ void fill_zero(float* __restrict__ p, size_t n) {
  size_t i = (size_t)blockIdx.x * blockDim.x + threadIdx.x;
  if (i < n) p[i] = 0.f;
}

// out = acc + self-loop message + bias   (wave per node)
__global__ __launch_bounds__(256) void finalize_g(
    const float* __restrict__ acc, const float* __restrict__ h,
    const float* __restrict__ asrc, const float* __restrict__ adst,
    const float* __restrict__ amax, const float* __restrict__ den,
    const float* __restrict__ bias, float* __restrict__ out, int N) {
  const int n = blockIdx.x * (blockDim.x >> 5) + (threadIdx.x >> 5);
  if (n >= N) return;
  const int lane = threadIdx.x & 31;
  const int hd = lane >> 3;
  const float al = lrelu(asrc[n * NHEAD + hd] + adst[n * NHEAD + hd]);
  const float w  = __expf(al - amax[n * NHEAD + hd]) / den[n * NHEAD + hd];
  const float4 hv = ((const float4*)(h   + (size_t)n * HC))[lane];
  const float4 av = ((const float4*)(acc + (size_t)n * HC))[lane];
  const float4 bv = ((const float4*)bias)[lane];
  float4 o;
  o.x = av.x + hv.x * w + bv.x;
  o.y = av.y + hv.y * w + bv.y;
  o.z = av.z + hv.z * w + bv.z;
  o.w = av.w + hv.w * w + bv.w;
  ((float4*)(out + (size_t)n * HC))[lane] = o;
}

// x1_combined = gat1(n) + b1 + x2_out[group[n]]   (x2_out already has b2)
__global__ __launch_bounds__(256) void finalize_combined(
    const float* __restrict__ acc, const float* __restrict__ h,
    const float* __restrict__ asrc, const float* __restrict__ adst,
    const float* __restrict__ amax, const float* __restrict__ den,
    const float* __restrict__ bias, const long long* __restrict__ grp,
    const float* __restrict__ out2, float* __restrict__ out, int N) {
  const int n = blockIdx.x * (blockDim.x >> 5) + (threadIdx.x >> 5);
  if (n >= N) return;
  const int lane = threadIdx.x & 31;
  const int hd = lane >> 3;
  const float al = lrelu(asrc[n * NHEAD + hd] + adst[n * NHEAD + hd]);
  const float w  = __expf(al - amax[n * NHEAD + hd]) / den[n * NHEAD + hd];
  const int g = (int)grp[n];
  const float4 hv = ((const float4*)(h    + (size_t)n * HC))[lane];
  const float4 av = ((const float4*)(acc  + (size_t)n * HC))[lane];
  const float4 bv = ((const float4*)bias)[lane];
  const float4 gv = ((const float4*)(out2 + (size_t)g * HC))[lane];
  float4 o;
  o.x = av.x + hv.x * w + bv.x + gv.x;
  o.y = av.y + hv.y * w + bv.y + gv.y;
  o.z = av.z + hv.z * w + bv.z + gv.z;
  o.w = av.w + hv.w * w + bv.w + gv.w;
  ((float4*)(out + (size_t)n * HC))[lane] = o;
}

// ------------------------------------------------------------------ host ----
static void run_gat(const float* X, const long long* EI, const float* W,
                    const float* att_s, const float* att_d,
                    int N, int E,
                    float* h, float* acc, float* asrc, float* adst,
                    float* amax, float* den, hipStream_t stream) {
  gemm_xw_wmma<<<(N + 15) / 16, 256, 0, stream>>>(X, W, h, N);
  fill_zero<<<(unsigned)(((size_t)N * HC + 255) / 256), 256, 0, stream>>>(
      acc, (size_t)N * HC);
  node_att<<<(N * NHEAD + 255) / 256, 256, 0, stream>>>(
      h, att_s, att_d, asrc, adst, amax, N);
  edge_amax<<<(E * NHEAD + 255) / 256, 256, 0, stream>>>(EI, asrc, adst, amax, E);
  node_den<<<(N * NHEAD + 255) / 256, 256, 0, stream>>>(asrc, adst, amax, den, N);
  edge_den<<<(E * NHEAD + 255) / 256, 256, 0, stream>>>(
      EI, asrc, adst, amax, den, E);
  edge_aggregate<<<(E + 7) / 8, 256, 0, stream>>>(
      EI, h, asrc, adst, amax, den, acc, E);
}

extern "C" void kernel_launch(void* const* d_in, const int* in_sizes, int n_in,
                              void* d_out, int out_size, void* d_ws, size_t ws_size,
                              hipStream_t stream) {
  (void)n_in; (void)out_size; (void)ws_size;
  const float*     x1  = (const float*)d_in[0];
  const long long* ei1 = (const long long*)d_in[1];
  const float*     x2  = (const float*)d_in[2];
  const long long* ei2 = (const long long*)d_in[3];
  const long long* grp = (const long long*)d_in[4];
  const float*     W1  = (const float*)d_in[5];
  const float*     as1 = (const float*)d_in[6];
  const float*     ad1 = (const float*)d_in[7];
  const float*     b1  = (const float*)d_in[8];
  const float*     W2  = (const float*)d_in[9];
  const float*     as2 = (const float*)d_in[10];
  const float*     ad2 = (const float*)d_in[11];
  const float*     b2  = (const float*)d_in[12];

  const int N1 = in_sizes[0] / IN_F;
  const int E1 = in_sizes[1] / 2;
  const int N2 = in_sizes[2] / IN_F;
  const int E2 = in_sizes[3] / 2;

  // Workspace carve-up (~65 MB for the reference sizes).
  float* ws = (float*)d_ws;
  float* h1    = ws; ws += (size_t)N1 * HC;
  float* h2    = ws; ws += (size_t)N2 * HC;
  float* acc1  = ws; ws += (size_t)N1 * HC;
  float* acc2  = ws; ws += (size_t)N2 * HC;
  float* asrc1 = ws; ws += (size_t)N1 * NHEAD;
  float* adst1 = ws; ws += (size_t)N1 * NHEAD;
  float* amax1 = ws; ws += (size_t)N1 * NHEAD;
  float* den1  = ws; ws += (size_t)N1 * NHEAD;
  float* asrc2 = ws; ws += (size_t)N2 * NHEAD;
  float* adst2 = ws; ws += (size_t)N2 * NHEAD;
  float* amax2 = ws; ws += (size_t)N2 * NHEAD;
  float* den2  = ws; ws += (size_t)N2 * NHEAD;

  run_gat(x1, ei1, W1, as1, ad1, N1, E1, h1, acc1, asrc1, adst1, amax1, den1, stream);
  run_gat(x2, ei2, W2, as2, ad2, N2, E2, h2, acc2, asrc2, adst2, amax2, den2, stream);

  float* out1 = (float*)d_out;                 // [N1,128]
  float* out2 = out1 + (size_t)N1 * HC;        // [N2,128]

  // x2_out (with bias) first, then the combined x1 output that gathers it.
  finalize_g<<<(N2 + 7) / 8, 256, 0, stream>>>(
      acc2, h2, asrc2, adst2, amax2, den2, b2, out2, N2);
  finalize_combined<<<(N1 + 7) / 8, 256, 0, stream>>>(
      acc1, h1, asrc1, adst1, amax1, den1, b1, grp, out2, out1, N1);
}